// CMSWrite_37391985279471
// MI455X (gfx1250) — compile-verified
//
#include <hip/hip_runtime.h>
#include <math.h>

#define L_LVL 3
#define N_SLOT 65536
#define D_S 1024
#define D_E 512
#define D_C 256
#define D_K 128
#define D_Z 128
#define D_LVL 512
#define D_OUT 640   // D_LVL + D_K

typedef float v2f __attribute__((ext_vector_type(2)));
typedef float v4f __attribute__((ext_vector_type(4)));
typedef float v8f __attribute__((ext_vector_type(8)));

__device__ __forceinline__ float wave_sum(float v) {
#pragma unroll
    for (int off = 16; off > 0; off >>= 1) v += __shfl_xor(v, off, 32);
    return v;
}
__device__ __forceinline__ float wave_max(float v) {
#pragma unroll
    for (int off = 16; off > 0; off >>= 1) v = fmaxf(v, __shfl_xor(v, off, 32));
    return v;
}

// ---------------------------------------------------------------------------
// Kernel 1: z (matvec + layernorm + relu), gate (with threshold), v, k.
// One block PER LEVEL (levels are independent) to cut serial latency 3x.
// All heavy reductions are wave-level shuffle reduces with coalesced reads.
// ---------------------------------------------------------------------------
__global__ __launch_bounds__(256) void prep_kernel(
    const float* __restrict__ s_t, const float* __restrict__ e_t,
    const float* __restrict__ lc,
    const float* __restrict__ W_ev0, const float* __restrict__ W_ev,
    const float* __restrict__ b_ev,
    const float* __restrict__ ln_g, const float* __restrict__ ln_b,
    const float* __restrict__ W_gate, const float* __restrict__ b_gate,
    const float* __restrict__ W_val, const float* __restrict__ b_val,
    const float* __restrict__ W_key, const float* __restrict__ b_key,
    float* __restrict__ kvec, float* __restrict__ vvec, float* __restrict__ gbuf)
{
    __shared__ float ev[D_S + D_C + D_E];   // 1792
    __shared__ float zb[D_Z];
    __shared__ float musig[2];

    const int t = threadIdx.x, lane = t & 31, wave = t >> 5;
    const int l = blockIdx.x;

    const int dim = (l == 0) ? (D_S + D_E) : (D_S + D_C + D_E);
    // Build concatenated event vector in LDS.
    for (int i = t; i < D_S; i += 256) ev[i] = s_t[i];
    if (l == 0) {
        for (int i = t; i < D_E; i += 256) ev[D_S + i] = e_t[i];
    } else {
        for (int i = t; i < D_C; i += 256) ev[D_S + i] = lc[(l - 1) * D_C + i];
        for (int i = t; i < D_E; i += 256) ev[D_S + D_C + i] = e_t[i];
    }
    __syncthreads();

    const float* W = (l == 0) ? W_ev0
                              : (W_ev + (size_t)(l - 1) * D_Z * (D_S + D_C + D_E));
    // z_pre rows: wave-per-row, 16 rows per wave.
    for (int j = wave; j < D_Z; j += 8) {
        const float* wr = W + (size_t)j * dim;
        float p = 0.f;
        for (int i = lane; i < dim; i += 32) p += wr[i] * ev[i];
        p = wave_sum(p);
        if (lane == 0) zb[j] = p + b_ev[l * D_Z + j];
    }
    __syncthreads();

    // LayerNorm stats on wave 0.
    if (wave == 0) {
        float s = 0.f, ss = 0.f;
#pragma unroll
        for (int r = 0; r < 4; ++r) {
            float x = zb[lane + 32 * r];
            s += x; ss += x * x;
        }
        s = wave_sum(s); ss = wave_sum(ss);
        if (lane == 0) {
            float mu  = s * (1.0f / D_Z);
            float var = ss * (1.0f / D_Z) - mu * mu;
            musig[0] = mu;
            musig[1] = rsqrtf(var + 1e-5f);
        }
    }
    __syncthreads();

    if (t < D_Z) {
        float zn = (zb[t] - musig[0]) * musig[1] * ln_g[l * D_Z + t] + ln_b[l * D_Z + t];
        zb[t] = zn > 0.f ? zn : 0.f;   // relu
    }
    __syncthreads();

    // Gate (wave 0) with sparse-write threshold.
    if (wave == 0) {
        float p = 0.f;
#pragma unroll
        for (int r = 0; r < 4; ++r)
            p += zb[lane + 32 * r] * W_gate[l * D_Z + lane + 32 * r];
        p = wave_sum(p);
        if (lane == 0) {
            float gg = 1.0f / (1.0f + expf(-(p + b_gate[l])));
            gbuf[l] = (gg < 0.1f) ? 0.0f : gg;
        }
    }
    // v = tanh(W_val z + b)   (512 rows, wave-per-row)
    for (int r = wave; r < D_LVL; r += 8) {
        const float* wr = W_val + ((size_t)l * D_LVL + r) * D_Z;
        float p = 0.f;
#pragma unroll
        for (int i = lane; i < D_Z; i += 32) p += wr[i] * zb[i];
        p = wave_sum(p);
        if (lane == 0) vvec[l * D_LVL + r] = tanhf(p + b_val[l * D_LVL + r]);
    }
    // k = W_key z + b         (128 rows)
    for (int r = wave; r < D_K; r += 8) {
        const float* wr = W_key + ((size_t)l * D_K + r) * D_Z;
        float p = 0.f;
#pragma unroll
        for (int i = lane; i < D_Z; i += 32) p += wr[i] * zb[i];
        p = wave_sum(p);
        if (lane == 0) kvec[l * D_K + r] = p + b_key[l * D_K + r];
    }
}

// ---------------------------------------------------------------------------
// Kernel 2: scores[l][n] = dot(K[l,n,:], k[l,:]) / sqrt(128) via
// V_WMMA_F32_16X16X4_F32. Each wave owns 16 rows of K; 32 accumulating
// WMMAs cover K=128. B broadcasts the 4-chunk of k to all 16 columns, so
// column 0 of D holds exact fp32 scores.
//   A layout (ISA 16x4 f32): lanes 0-15 -> K=0,1 ; lanes 16-31 -> K=2,3.
//   D layout: lane 0 vgpr r = D[r][0]; lane 16 vgpr r = D[8+r][0].
// K reads are non-temporal: 101 MB cannot stay L2-resident until the update
// pass (>900 MB streams through in between), so don't pollute L2 with it.
// ---------------------------------------------------------------------------
__global__ __launch_bounds__(256) void scores_kernel(
    const float* __restrict__ Kmat,   // [L, N, D_K]
    const float* __restrict__ kvec,   // [L, D_K]
    float* __restrict__ scores)       // [L, N]
{
    const int wave = threadIdx.x >> 5;
    const int lane = threadIdx.x & 31;
    const int blocksPerLevel = N_SLOT / 128;               // 512
    const int lvl = blockIdx.x / blocksPerLevel;
    const int n0  = (blockIdx.x % blocksPerLevel) * 128 + wave * 16;

    const float* __restrict__ Kbase = Kmat + ((size_t)lvl * N_SLOT + n0) * D_K;
    const float* __restrict__ kv    = kvec + lvl * D_K;

    const int m  = lane & 15;            // matrix row within 16-row tile
    const int kh = (lane >> 4) << 1;     // 0 for lanes 0-15, 2 for lanes 16-31
    const float* __restrict__ arow = Kbase + (size_t)m * D_K + kh;

    v8f acc = {};
#pragma unroll 8
    for (int kc = 0; kc < D_K; kc += 4) {
        v2f a = __builtin_nontemporal_load(reinterpret_cast<const v2f*>(arow + kc));
        v2f b = *reinterpret_cast<const v2f*>(kv + kc + kh);
        acc = __builtin_amdgcn_wmma_f32_16x16x4_f32(
            false, a, false, b, (short)0, acc, false, false);
    }

    const float scale = 0.08838834764831845f;  // 1/sqrt(128)
    if ((lane & 15) == 0) {
        float* sp = scores + (size_t)lvl * N_SLOT + n0 + (lane >> 4) * 8;
#pragma unroll
        for (int r = 0; r < 8; ++r) sp[r] = acc[r] * scale;
    }
}

// ---------------------------------------------------------------------------
// Kernel 3: per-level softmax stats (max, 1/sum(exp)). One block per level;
// scores (786 KB total) are L2-resident after kernel 2.
// ---------------------------------------------------------------------------
__global__ __launch_bounds__(256) void stats_kernel(
    const float* __restrict__ scores, float* __restrict__ stats)
{
    const int l = blockIdx.x;
    const float* __restrict__ s = scores + (size_t)l * N_SLOT;
    __shared__ float redm[8];
    __shared__ float reds[8];
    const int t = threadIdx.x, lane = t & 31, wave = t >> 5;

    float mx = -1e30f;
    for (int i = t; i < N_SLOT; i += 256) mx = fmaxf(mx, s[i]);
    mx = wave_max(mx);
    if (lane == 0) redm[wave] = mx;
    __syncthreads();
    if (t == 0) {
        float m = redm[0];
#pragma unroll
        for (int i = 1; i < 8; ++i) m = fmaxf(m, redm[i]);
        redm[0] = m;
    }
    __syncthreads();
    mx = redm[0];

    float sum = 0.f;
    for (int i = t; i < N_SLOT; i += 256) sum += expf(s[i] - mx);
    sum = wave_sum(sum);
    if (lane == 0) reds[wave] = sum;
    __syncthreads();
    if (t == 0) {
        float S = 0.f;
#pragma unroll
        for (int i = 0; i < 8; ++i) S += reds[i];
        stats[l * 2 + 0] = mx;
        stats[l * 2 + 1] = 1.0f / S;
    }
}

// ---------------------------------------------------------------------------
// Kernel 4: bandwidth pass (~1 GB of read-once/write-once streaming).
// out[l,n,:] = [ keep*M[l,n,:] + ga*v[l,:] , keep*K[l,n,:] + ga*k[l,:] ]
// 8 rows per block, v4f (B128) loads/stores, v/k cached in LDS.
// All bulk traffic is non-temporal (TH=NT): nothing here is re-used, and
// regular-temporal traffic would just thrash the 192 MB L2.
// ---------------------------------------------------------------------------
__global__ __launch_bounds__(256) void update_kernel(
    const float* __restrict__ Mmat,    // [L, N, 512]
    const float* __restrict__ Kmat,    // [L, N, 128]
    const float* __restrict__ scores,  // [L, N]
    const float* __restrict__ vvec,    // [L, 512]
    const float* __restrict__ kvec,    // [L, 128]
    const float* __restrict__ gbuf,    // [L]
    const float* __restrict__ stats,   // [L, 2]
    const float* __restrict__ decay,   // [L]
    float* __restrict__ out)           // [L, N, 640]
{
    const int ROWS = 8;
    const int blocksPerLevel = N_SLOT / ROWS;
    const int lvl = blockIdx.x / blocksPerLevel;
    const int n0  = (blockIdx.x % blocksPerLevel) * ROWS;

    __shared__ __align__(16) float vsh[D_LVL];
    __shared__ __align__(16) float ksh[D_K];
    __shared__ float ga[ROWS];
    __shared__ float keepv;

    const int t = threadIdx.x;
    for (int i = t; i < D_LVL; i += 256) vsh[i] = vvec[lvl * D_LVL + i];
    if (t < D_K) ksh[t] = kvec[lvl * D_K + t];
    if (t < ROWS) {
        float mx  = stats[lvl * 2 + 0];
        float inv = stats[lvl * 2 + 1];
        ga[t] = gbuf[lvl] * expf(scores[(size_t)lvl * N_SLOT + n0 + t] - mx) * inv;
    }
    if (t == 0) keepv = 1.0f - decay[lvl];
    __syncthreads();

    const float kp = keepv;
    const size_t rowBase = (size_t)lvl * N_SLOT + n0;
    const v4f* __restrict__ Mp = (const v4f*)(Mmat + rowBase * D_LVL);
    const v4f* __restrict__ Kp = (const v4f*)(Kmat + rowBase * D_K);
    v4f* __restrict__ Op = (v4f*)(out + rowBase * D_OUT);

    // 8 rows * 160 v4f per row = 1280 v4f -> 5 per thread.
#pragma unroll
    for (int it = 0; it < 5; ++it) {
        int idx = t + it * 256;
        int r = idx / 160;
        int c = idx - r * 160;
        float gar = ga[r];
        v4f o;
        if (c < 128) {
            v4f m4 = __builtin_nontemporal_load(Mp + (size_t)r * 128 + c);
            v4f vv = *reinterpret_cast<const v4f*>(&vsh[c * 4]);
            o = kp * m4 + gar * vv;
        } else {
            int cc = c - 128;
            v4f k4 = __builtin_nontemporal_load(Kp + (size_t)r * 32 + cc);
            v4f kk = *reinterpret_cast<const v4f*>(&ksh[cc * 4]);
            o = kp * k4 + gar * kk;
        }
        __builtin_nontemporal_store(o, Op + (size_t)r * 160 + c);
    }
}

// ---------------------------------------------------------------------------
extern "C" void kernel_launch(void* const* d_in, const int* in_sizes, int n_in,
                              void* d_out, int out_size, void* d_ws, size_t ws_size,
                              hipStream_t stream) {
    const float* s_t    = (const float*)d_in[0];
    const float* e_t    = (const float*)d_in[1];
    const float* lc     = (const float*)d_in[2];
    const float* W_ev0  = (const float*)d_in[3];
    const float* W_ev   = (const float*)d_in[4];
    const float* b_ev   = (const float*)d_in[5];
    const float* ln_g   = (const float*)d_in[6];
    const float* ln_b   = (const float*)d_in[7];
    const float* W_gate = (const float*)d_in[8];
    const float* b_gate = (const float*)d_in[9];
    const float* W_val  = (const float*)d_in[10];
    const float* b_val  = (const float*)d_in[11];
    const float* W_key  = (const float*)d_in[12];
    const float* b_key  = (const float*)d_in[13];
    const float* Mmat   = (const float*)d_in[14];
    const float* Kmat   = (const float*)d_in[15];
    const float* decay  = (const float*)d_in[16];
    float* out = (float*)d_out;

    // Workspace layout (floats): kvec[384] | vvec[1536] | gbuf[3] | stats[6]
    //                            | pad | scores[L*N]
    float* ws     = (float*)d_ws;
    float* kvec   = ws;                 // L*128
    float* vvec   = ws + 384;           // L*512
    float* gbuf   = ws + 1920;          // L
    float* stats  = ws + 1924;          // L*2
    float* scores = ws + 1936;          // L*N = 196608

    prep_kernel<<<L_LVL, 256, 0, stream>>>(s_t, e_t, lc, W_ev0, W_ev, b_ev,
                                           ln_g, ln_b, W_gate, b_gate,
                                           W_val, b_val, W_key, b_key,
                                           kvec, vvec, gbuf);

    scores_kernel<<<L_LVL * (N_SLOT / 128), 256, 0, stream>>>(Kmat, kvec, scores);

    stats_kernel<<<L_LVL, 256, 0, stream>>>(scores, stats);

    update_kernel<<<L_LVL * (N_SLOT / 8), 256, 0, stream>>>(
        Mmat, Kmat, scores, vvec, kvec, gbuf, stats, decay, out);
}